// LatticeEmbeddedNet_9569187136212
// MI455X (gfx1250) — compile-verified
//
#include <hip/hip_runtime.h>

typedef __attribute__((ext_vector_type(2))) float v2f;
typedef __attribute__((ext_vector_type(8))) float v8f;

#define SIGMA_SQ   (0.5f * 0.5f)
#define BUMP_EPS   0.001f

// One block per 16-row eval tile. 8 waves/block, each wave strides over
// 16-column site tiles. Cross term e.s computed with V_WMMA_F32_16X16X4_F32
// (K padded 3->4 with zeros), epilogue exp/relu/weighted-sum in VALU.
__global__ __launch_bounds__(256)
void lattice_bump_wmma_kernel(const float* __restrict__ eval_points,  // (n_eval,3)
                              const float* __restrict__ site_coords,  // (n_sites,3)
                              const float* __restrict__ weights,      // (n_sites,)
                              float* __restrict__ out,                // (n_eval,)
                              int n_eval, int n_sites)
{
    const int lane   = threadIdx.x & 31;
    const int wave   = threadIdx.x >> 5;
    const int nwaves = blockDim.x >> 5;
    const int tile_m = blockIdx.x;
    const int half   = lane >> 4;     // 0: K=0,1 ; 1: K=2,3
    const int lid    = lane & 15;     // row (for A) / column (for B,D)

    // ---- A matrix: 16x4 eval tile, f32, 2 VGPRs ----
    // lanes 0-15: (x, y) of row lid ; lanes 16-31: (z, 0) of row lid
    const int m0 = tile_m * 16 + lid;
    const float ex = eval_points[3 * m0 + 0];
    const float ey = eval_points[3 * m0 + 1];
    const float ez = eval_points[3 * m0 + 2];
    v2f a;
    a.x = half ? ez : ex;
    a.y = half ? 0.0f : ey;

    // ---- |e|^2 per D-row r: lanes 0-15 need row r, lanes 16-31 row r+8 ----
    float e2v[8];
#pragma unroll
    for (int r = 0; r < 8; ++r) {
        const int mm = tile_m * 16 + r + half * 8;
        const float x = eval_points[3 * mm + 0];
        const float y = eval_points[3 * mm + 1];
        const float z = eval_points[3 * mm + 2];
        e2v[r] = x * x + y * y + z * z;
    }

    const float inv_s2 = 1.0f / SIGMA_SQ;
    float acc[8];
#pragma unroll
    for (int r = 0; r < 8; ++r) acc[r] = 0.0f;

    const int ntiles = n_sites >> 4;   // site tiles of 16
    for (int t = wave; t < ntiles; t += nwaves) {
        const int n = t * 16 + lid;    // this lane's column (site index)

        // prefetch next tile this wave will touch (global_prefetch_b8)
        if (t + nwaves < ntiles)
            __builtin_prefetch(site_coords + 3 * (n + 16 * nwaves), 0, 0);

        const float sx = site_coords[3 * n + 0];
        const float sy = site_coords[3 * n + 1];
        const float sz = site_coords[3 * n + 2];

        // ---- B matrix: 4x16 site tile, f32, 2 VGPRs (mirror of A) ----
        v2f b;
        b.x = half ? sz : sx;
        b.y = half ? 0.0f : sy;

        const float s2 = sx * sx + sy * sy + sz * sz;
        const float w  = weights[n];

        v8f c = {};
        c = __builtin_amdgcn_wmma_f32_16x16x4_f32(
                /*neg_a=*/false, a, /*neg_b=*/false, b,
                /*c_mod=*/(short)0, c, /*reuse_a=*/false, /*reuse_b=*/false);

        // epilogue: r2 -> exp -> relu -> weighted accumulate over columns
#pragma unroll
        for (int r = 0; r < 8; ++r) {
            const float r2  = (e2v[r] + s2 - 2.0f * c[r]) * inv_s2;
            const float g   = __expf(-0.5f * r2) - BUMP_EPS;
            const float phi = fmaxf(g, 0.0f);
            acc[r] += phi * w;
        }
    }

    // ---- reduce across the 16 columns (xor offsets stay inside each half) ----
#pragma unroll
    for (int r = 0; r < 8; ++r) {
        acc[r] += __shfl_xor(acc[r], 1, 32);
        acc[r] += __shfl_xor(acc[r], 2, 32);
        acc[r] += __shfl_xor(acc[r], 4, 32);
        acc[r] += __shfl_xor(acc[r], 8, 32);
    }

    // ---- cross-wave reduction in LDS ----
    __shared__ float red[8][16];
    if (lid == 0) {                     // lanes 0 and 16 of each wave
#pragma unroll
        for (int r = 0; r < 8; ++r) red[wave][r + half * 8] = acc[r];
    }
    __syncthreads();

    if ((int)threadIdx.x < 16) {
        float s = 0.0f;
        for (int w2 = 0; w2 < nwaves; ++w2) s += red[w2][threadIdx.x];
        const int mg = tile_m * 16 + threadIdx.x;
        if (mg < n_eval) out[mg] = s;
    }
}

extern "C" void kernel_launch(void* const* d_in, const int* in_sizes, int n_in,
                              void* d_out, int out_size, void* d_ws, size_t ws_size,
                              hipStream_t stream) {
    const float* eval_points = (const float*)d_in[0];
    const float* site_coords = (const float*)d_in[1];
    const float* weights     = (const float*)d_in[2];
    float* out = (float*)d_out;

    const int n_eval  = in_sizes[0] / 3;   // (n_eval, 3)
    const int n_sites = in_sizes[2];       // (n_sites,)

    const int grid = (n_eval + 15) / 16;   // one block per 16-row eval tile
    lattice_bump_wmma_kernel<<<grid, 256, 0, stream>>>(
        eval_points, site_coords, weights, out, n_eval, n_sites);
}